// Length_Regulator_4277787427515
// MI455X (gfx1250) — compile-verified
//
#include <hip/hip_runtime.h>
#include <stdint.h>

// Problem constants (from the reference).
#define B_      32
#define C_      512
#define T_IN    1024
#define T_OUT   4096
#define C_CHUNK 16     // channels per gather block

// Native clang vector type: required by __builtin_nontemporal_store
// (HIP_vector_type<float,4> is a class and is rejected).
typedef float v4f __attribute__((ext_vector_type(4)));

// ---------------------------------------------------------------------------
// Kernel 1: per batch row b, inclusive scan of durations, then scatter-fill
// the index map: thread j writes j into idx[b, cum_excl(j) .. cum_incl(j)).
// Wave32-native scan: __shfl_up within each wave (no barriers), one LDS
// exchange of the 32 wave sums, one more shuffle scan. 2 barriers total.
// One block per batch row, 1024 threads (32 wave32 waves).
// ---------------------------------------------------------------------------
__global__ __launch_bounds__(T_IN) void lr_scan_idx(const int* __restrict__ dur,
                                                    int* __restrict__ idx) {
  const int b    = blockIdx.x;
  const int tid  = threadIdx.x;
  const int lane = tid & 31;
  const int wave = tid >> 5;

  __shared__ int wsum[32];

  const int v = dur[b * T_IN + tid];

  // Per-wave inclusive scan (wave32: 5 shuffle steps).
  int s = v;
#pragma unroll
  for (int d = 1; d < 32; d <<= 1) {
    const int t = __shfl_up(s, d, 32);
    if (lane >= d) s += t;
  }
  if (lane == 31) wsum[wave] = s;
  __syncthreads();

  // Wave 0 scans the 32 wave totals.
  if (wave == 0) {
    int w = wsum[lane];
#pragma unroll
    for (int d = 1; d < 32; d <<= 1) {
      const int t = __shfl_up(w, d, 32);
      if (lane >= d) w += t;
    }
    wsum[lane] = w;
  }
  __syncthreads();

  const int base = (wave > 0) ? wsum[wave - 1] : 0;
  const int incl = base + s;       // cumsum inclusive at j = tid
  const int excl = incl - v;       // cumsum exclusive

  // Scatter: idx[b, t] = tid for t in [excl, incl). Covers all of [0, T_OUT)
  // since cum[T_IN-1] == T_OUT; matches searchsorted(cum, t, 'right').
  int* op = idx + (size_t)b * T_OUT;
  for (int p = excl; p < incl; ++p) op[p] = tid;
}

// ---------------------------------------------------------------------------
// Async global->LDS b128 copy (CDNA5 async path, tracked by ASYNCcnt).
// lds_addr: raw LDS byte address (low 32 bits of the generic pointer),
// gaddr:    64-bit global address.
// ---------------------------------------------------------------------------
__device__ __forceinline__ void async_b128_to_lds(uint32_t lds_addr,
                                                  const void* gaddr) {
  asm volatile("global_load_async_to_lds_b128 %0, %1, off"
               :
               : "v"(lds_addr), "v"((uint64_t)(uintptr_t)gaddr)
               : "memory");
}

// ---------------------------------------------------------------------------
// Kernel 2: gather. Each block owns one batch row b and C_CHUNK channels,
// covering the full T_OUT, so each 4KB encoding row is staged into LDS
// exactly once device-wide (one async b128 per lane, double-buffered across
// channels). The monotone gather is served from LDS (same-address reads
// broadcast), and the 256MB output is written with coalesced non-temporal
// b128 stores (th:TH_STORE_NT - written once, never re-read).
// Block = 256 threads = 8 wave32 waves. Grid = (C_/C_CHUNK, B_) = 1024 blocks.
// ---------------------------------------------------------------------------
__global__ __launch_bounds__(256) void lr_gather(const float* __restrict__ enc,
                                                 const int*  __restrict__ idx,
                                                 float*      __restrict__ out) {
  const int b   = blockIdx.y;
  const int c0  = blockIdx.x * C_CHUNK;
  const int tid = threadIdx.x;

  __shared__ float s_row[2][T_IN];   // 8KB double buffer

  // Per-thread gather indices (constant across all channels of this block).
  // Thread tid handles output elements [4*(tid+k*256) .. +3] for k=0..3.
  int myidx[16];
  const int4* ip = (const int4*)(idx + (size_t)b * T_OUT);
#pragma unroll
  for (int k = 0; k < 4; ++k) {
    const int4 v = ip[tid + k * 256];
    myidx[k * 4 + 0] = v.x;
    myidx[k * 4 + 1] = v.y;
    myidx[k * 4 + 2] = v.z;
    myidx[k * 4 + 3] = v.w;
  }

  const float* ebase = enc + ((size_t)b * C_ + c0) * T_IN;

  const uint32_t lds0 = (uint32_t)(uintptr_t)(&s_row[0][0]) + (uint32_t)tid * 16u;
  const uint32_t lds1 = (uint32_t)(uintptr_t)(&s_row[1][0]) + (uint32_t)tid * 16u;

  // Prologue: stage channel 0 into buffer 0 (256 lanes x 16B = 4KB row).
  async_b128_to_lds(lds0, (const char*)ebase + (size_t)tid * 16);

#pragma unroll 1
  for (int i = 0; i < C_CHUNK; ++i) {
    const int cur = i & 1;

    if (i + 1 < C_CHUNK) {
      // Stage next row into the other buffer, then wait until only that
      // new stage is outstanding: async loads complete in order, so the
      // current buffer's stage has landed in LDS.
      async_b128_to_lds(cur ? lds0 : lds1,
                        (const char*)(ebase + (size_t)(i + 1) * T_IN) +
                            (size_t)tid * 16);
      asm volatile("s_wait_asynccnt 1" ::: "memory");
    } else {
      asm volatile("s_wait_asynccnt 0" ::: "memory");
    }
    __syncthreads();  // all waves' slices of the current buffer are in LDS

    const float* r = s_row[cur];
    v4f* orow = (v4f*)(out + ((size_t)b * C_ + (size_t)(c0 + i)) * T_OUT);
#pragma unroll
    for (int k = 0; k < 4; ++k) {
      v4f v;
      v.x = r[myidx[k * 4 + 0]];
      v.y = r[myidx[k * 4 + 1]];
      v.z = r[myidx[k * 4 + 2]];
      v.w = r[myidx[k * 4 + 3]];
      __builtin_nontemporal_store(v, &orow[tid + k * 256]);  // b128 NT store
    }
    __syncthreads();  // everyone done reading s_row[cur] before it is restaged
  }
}

// ---------------------------------------------------------------------------
// Launch: kernel 1 fills d_ws with the [B, T_OUT] index map (512KB),
// kernel 2 streams the gather.
// ---------------------------------------------------------------------------
extern "C" void kernel_launch(void* const* d_in, const int* in_sizes, int n_in,
                              void* d_out, int out_size, void* d_ws,
                              size_t ws_size, hipStream_t stream) {
  const float* enc = (const float*)d_in[0];
  const int*   dur = (const int*)d_in[1];
  float*       out = (float*)d_out;
  int*         idx = (int*)d_ws;  // B_*T_OUT ints = 512KB

  lr_scan_idx<<<dim3(B_), dim3(T_IN), 0, stream>>>(dur, idx);
  lr_gather<<<dim3(C_ / C_CHUNK, B_), dim3(256), 0, stream>>>(enc, idx, out);
}